// DGCNNSTATE_42683384987988
// MI455X (gfx1250) — compile-verified
//
#include <hip/hip_runtime.h>
#include <hip/hip_bf16.h>
#include <math.h>

typedef __attribute__((ext_vector_type(2))) float v2f;
typedef __attribute__((ext_vector_type(8))) float v8f;

#define N_NODES   65536
#define NODES_PG  128
#define N_GRAPHS  512
#define N_EDGES   1048576
#define C_FEAT    64
#define K_POOL    35
#define CONV_OUT  32
#define CONV_K    4
#define CONV_L    32          /* K_POOL - CONV_K + 1 */
#define LIN1_IN   1024        /* CONV_OUT * CONV_L */
#define N_CLASSES 10

/* ---------------- generic zero fill ---------------- */
__global__ void zero_f32(float* __restrict__ p, int n) {
    int i = blockIdx.x * blockDim.x + threadIdx.x;
    int stride = gridDim.x * blockDim.x;
    for (; i < n; i += stride) p[i] = 0.0f;
}

/* ---------------- degree count + reciprocal ---------------- */
__global__ void count_edges(const int* __restrict__ dst, float* __restrict__ cnt) {
    int e = blockIdx.x * blockDim.x + threadIdx.x;
    if (e < N_EDGES) atomicAdd(&cnt[dst[e]], 1.0f);
}

__global__ void finalize_rcnt(float* __restrict__ cnt) {
    int i = blockIdx.x * blockDim.x + threadIdx.x;
    if (i < N_NODES) cnt[i] = 1.0f / fmaxf(cnt[i], 1.0f);
}

/* ---------------- edge scatter: agg[dst] += h[src] ----------------
   one thread per (edge, feature) -> fully coalesced row reads,
   global_atomic_add_f32 into L2-resident agg                       */
__global__ void scatter_add(const float* __restrict__ hin,
                            const int* __restrict__ src,
                            const int* __restrict__ dst,
                            float* __restrict__ agg) {
    long long tid = (long long)blockIdx.x * blockDim.x + threadIdx.x;
    int e = (int)(tid >> 6);
    int f = (int)(tid & 63);
    if (e < N_EDGES) {
        float v = hin[(long long)src[e] * C_FEAT + f];
        atomicAdd(&agg[(long long)dst[e] * C_FEAT + f], v);
    }
}

/* ---------------- fused SAGE GEMM:
   out = relu( (agg*rcnt) @ Wl + h @ Wr + b )
   wave computes 16 rows x 64 cols with V_WMMA_F32_16X16X4_F32 ------ */
__global__ __launch_bounds__(256) void sage_gemm(
        const float* __restrict__ hin,   // [N,64]
        const float* __restrict__ agg,   // [N,64]
        const float* __restrict__ rcnt,  // [N]
        const float* __restrict__ Wl,    // [64,64] row-major (k,c)
        const float* __restrict__ Wr,    // [64,64]
        const float* __restrict__ bias,  // [64]
        float* __restrict__ hout)        // [N,64]
{
    __shared__ float sWl[64 * 64];
    __shared__ float sWr[64 * 64];
    int t = threadIdx.x;
    for (int i = t; i < 64 * 64; i += 256) { sWl[i] = Wl[i]; sWr[i] = Wr[i]; }
    __syncthreads();

    int wave = t >> 5;
    int lane = t & 31;
    int hi   = lane >> 4;     // which half of the wave
    int lm   = lane & 15;
    int row0 = blockIdx.x * 128 + wave * 16;
    int arow = row0 + lm;     // A-operand row for this lane
    float rc = rcnt[arow];

    v8f acc[4] = {};          // 4 N-tiles of 16 cols each

    for (int k0 = 0; k0 < C_FEAT; k0 += 4) {
        int ka = k0 + hi * 2; // f32 A 16x4 layout: lanes0-15 K=k0,k0+1; lanes16-31 K=k0+2,k0+3
        v2f am, ah;
        am.x = agg[(long long)arow * C_FEAT + ka]     * rc;
        am.y = agg[(long long)arow * C_FEAT + ka + 1] * rc;
        ah.x = hin[(long long)arow * C_FEAT + ka];
        ah.y = hin[(long long)arow * C_FEAT + ka + 1];
        for (int tn = 0; tn < 4; ++tn) {
            int col = tn * 16 + lm;
            v2f bl, br;       // B 4x16: VGPR j, half hi -> K=k0+2*hi+j, N=lm
            bl.x = sWl[(ka)     * 64 + col];
            bl.y = sWl[(ka + 1) * 64 + col];
            br.x = sWr[(ka)     * 64 + col];
            br.y = sWr[(ka + 1) * 64 + col];
            acc[tn] = __builtin_amdgcn_wmma_f32_16x16x4_f32(
                false, am, false, bl, (short)0, acc[tn], false, false);
            acc[tn] = __builtin_amdgcn_wmma_f32_16x16x4_f32(
                false, ah, false, br, (short)0, acc[tn], false, false);
        }
    }
    // C/D layout: VGPR v, lanes0-15 -> M=v, lanes16-31 -> M=v+8
    for (int tn = 0; tn < 4; ++tn) {
        int col = tn * 16 + lm;
        float b = bias[col];
        for (int v = 0; v < 8; ++v) {
            int m = v + hi * 8;
            float val = acc[tn][v] + b;
            hout[(long long)(row0 + m) * C_FEAT + col] = fmaxf(val, 0.0f);
        }
    }
}

/* ---------------- sort pooling: per-graph bitonic sort (desc) on ch63,
   take top-35, write transposed [G, 64, 35] ---------------- */
__global__ __launch_bounds__(128) void sortpool(const float* __restrict__ h,
                                                float* __restrict__ pooledT) {
    __shared__ float key[NODES_PG];
    __shared__ int   idx[NODES_PG];
    int g = blockIdx.x, t = threadIdx.x;
    key[t] = h[(long long)(g * NODES_PG + t) * C_FEAT + (C_FEAT - 1)];
    idx[t] = t;
    __syncthreads();
    for (int size = 2; size <= NODES_PG; size <<= 1) {
        for (int stride = size >> 1; stride > 0; stride >>= 1) {
            int j = t ^ stride;
            if (j > t) {
                bool up = ((t & size) == 0);   // descending overall
                float kt = key[t], kj = key[j];
                bool sw = up ? (kt < kj) : (kt > kj);
                if (sw) {
                    key[t] = kj; key[j] = kt;
                    int tmp = idx[t]; idx[t] = idx[j]; idx[j] = tmp;
                }
            }
            __syncthreads();
        }
    }
    for (int e = t; e < C_FEAT * K_POOL; e += NODES_PG) {
        int c = e / K_POOL, j = e % K_POOL;
        pooledT[((long long)g * C_FEAT + c) * K_POOL + j] =
            h[(long long)(g * NODES_PG + idx[j]) * C_FEAT + c];
    }
}

/* ---------------- conv1d (C=64 -> 32, k=4) + relu, flatten [G,1024] --- */
__global__ __launch_bounds__(256) void conv1d_relu(
        const float* __restrict__ pooledT,  // [G,64,35]
        const float* __restrict__ conv_w,   // [32,64,4]
        const float* __restrict__ conv_b,   // [32]
        float* __restrict__ yflat)          // [G,1024] (o*32+p)
{
    __shared__ float tile[C_FEAT * K_POOL];
    int g = blockIdx.x, t = threadIdx.x;
    for (int i = t; i < C_FEAT * K_POOL; i += 256)
        tile[i] = pooledT[(long long)g * C_FEAT * K_POOL + i];
    __syncthreads();
    for (int e = t; e < CONV_OUT * CONV_L; e += 256) {
        int o = e >> 5, p = e & 31;
        float s = conv_b[o];
        for (int c = 0; c < C_FEAT; ++c) {
            const float* w = &conv_w[(o * C_FEAT + c) * CONV_K];
            const float* x = &tile[c * K_POOL + p];
            s += w[0] * x[0] + w[1] * x[1] + w[2] * x[2] + w[3] * x[3];
        }
        yflat[(long long)g * LIN1_IN + e] = fmaxf(s, 0.0f);
    }
}

/* ---------------- generic WMMA GEMM: out = relu(A[Mx K] @ W[K x 64] + b) */
__global__ __launch_bounds__(256) void lin_gemm_relu(
        const float* __restrict__ A, const float* __restrict__ W,
        const float* __restrict__ bias, float* __restrict__ out, int K) {
    int t = threadIdx.x;
    int wave = t >> 5, lane = t & 31;
    int hi = lane >> 4, lm = lane & 15;
    int row0 = (blockIdx.x * 8 + wave) * 16;
    int arow = row0 + lm;
    v8f acc[4] = {};
    for (int k0 = 0; k0 < K; k0 += 4) {
        int ka = k0 + hi * 2;
        v2f a;
        a.x = A[(long long)arow * K + ka];
        a.y = A[(long long)arow * K + ka + 1];
        for (int tn = 0; tn < 4; ++tn) {
            int col = tn * 16 + lm;
            v2f b;
            b.x = W[(long long)(ka)     * 64 + col];
            b.y = W[(long long)(ka + 1) * 64 + col];
            acc[tn] = __builtin_amdgcn_wmma_f32_16x16x4_f32(
                false, a, false, b, (short)0, acc[tn], false, false);
        }
    }
    for (int tn = 0; tn < 4; ++tn) {
        int col = tn * 16 + lm;
        float b = bias[col];
        for (int v = 0; v < 8; ++v) {
            int m = v + hi * 8;
            float val = acc[tn][v] + b;
            out[(long long)(row0 + m) * 64 + col] = fmaxf(val, 0.0f);
        }
    }
}

/* ---------------- lin2 + log_softmax, one thread per graph ---------- */
__global__ void lin2_lsm(const float* __restrict__ z1,    // [G,64]
                         const float* __restrict__ W,     // [64,10]
                         const float* __restrict__ b,     // [10]
                         float* __restrict__ out) {       // [G,10]
    int r = blockIdx.x * blockDim.x + threadIdx.x;
    if (r >= N_GRAPHS) return;
    float logits[N_CLASSES];
    float m = -1e30f;
    for (int c = 0; c < N_CLASSES; ++c) {
        float s = b[c];
        for (int k = 0; k < 64; ++k) s += z1[r * 64 + k] * W[k * N_CLASSES + c];
        logits[c] = s;
        m = fmaxf(m, s);
    }
    float sum = 0.0f;
    for (int c = 0; c < N_CLASSES; ++c) sum += expf(logits[c] - m);
    float lse = m + logf(sum);
    for (int c = 0; c < N_CLASSES; ++c) out[r * N_CLASSES + c] = logits[c] - lse;
}

extern "C" void kernel_launch(void* const* d_in, const int* in_sizes, int n_in,
                              void* d_out, int out_size, void* d_ws, size_t ws_size,
                              hipStream_t stream) {
    const float* x       = (const float*)d_in[0];
    const int*   eidx    = (const int*)d_in[1];
    const int*   src     = eidx;
    const int*   dst     = eidx + N_EDGES;
    /* d_in[2] = batch (unused: regular 128-node graphs) */
    const float* W1l     = (const float*)d_in[3];
    const float* W1r     = (const float*)d_in[4];
    const float* b1      = (const float*)d_in[5];
    const float* Wl      = (const float*)d_in[6];   // [3,64,64]
    const float* Wr      = (const float*)d_in[7];
    const float* bs      = (const float*)d_in[8];   // [3,64]
    const float* conv_w  = (const float*)d_in[9];
    const float* conv_b  = (const float*)d_in[10];
    const float* lin1_w  = (const float*)d_in[11];
    const float* lin1_b  = (const float*)d_in[12];
    const float* lin2_w  = (const float*)d_in[13];
    const float* lin2_b  = (const float*)d_in[14];
    float* out = (float*)d_out;

    /* workspace layout */
    char* ws = (char*)d_ws;
    size_t off = 0;
    float* cnt     = (float*)(ws + off); off += (size_t)N_NODES * 4;               // 256 KB
    float* agg     = (float*)(ws + off); off += (size_t)N_NODES * C_FEAT * 4;      // 16 MB
    float* hA      = (float*)(ws + off); off += (size_t)N_NODES * C_FEAT * 4;      // 16 MB
    float* hB      = (float*)(ws + off); off += (size_t)N_NODES * C_FEAT * 4;      // 16 MB
    float* pooledT = (float*)(ws + off); off += (size_t)N_GRAPHS * C_FEAT * K_POOL * 4;
    float* yflat   = (float*)(ws + off); off += (size_t)N_GRAPHS * LIN1_IN * 4;
    float* z1      = (float*)(ws + off); off += (size_t)N_GRAPHS * 64 * 4;

    /* degree -> reciprocal (shared by all layers) */
    zero_f32<<<256, 256, 0, stream>>>(cnt, N_NODES);
    count_edges<<<N_EDGES / 256, 256, 0, stream>>>(dst, cnt);
    finalize_rcnt<<<N_NODES / 256, 256, 0, stream>>>(cnt);

    const int aggN = N_NODES * C_FEAT;
    const int scatterBlocks = (N_EDGES * 64) / 256;   // thread per (edge, feature)
    const int gemmBlocks = N_NODES / 128;             // 8 waves x 16 rows

    /* layer 0: x -> hA */
    zero_f32<<<4096, 256, 0, stream>>>(agg, aggN);
    scatter_add<<<scatterBlocks, 256, 0, stream>>>(x, src, dst, agg);
    sage_gemm<<<gemmBlocks, 256, 0, stream>>>(x, agg, cnt, W1l, W1r, b1, hA);

    /* layers 1..3 ping-pong hA/hB */
    float* hin = hA; float* hout = hB;
    for (int L = 0; L < 3; ++L) {
        zero_f32<<<4096, 256, 0, stream>>>(agg, aggN);
        scatter_add<<<scatterBlocks, 256, 0, stream>>>(hin, src, dst, agg);
        sage_gemm<<<gemmBlocks, 256, 0, stream>>>(
            hin, agg, cnt,
            Wl + (size_t)L * 64 * 64, Wr + (size_t)L * 64 * 64, bs + (size_t)L * 64,
            hout);
        float* tmp = hin; hin = hout; hout = tmp;
    }
    /* after 4 layers, result is in `hin` */

    sortpool<<<N_GRAPHS, NODES_PG, 0, stream>>>(hin, pooledT);
    conv1d_relu<<<N_GRAPHS, 256, 0, stream>>>(pooledT, conv_w, conv_b, yflat);
    lin_gemm_relu<<<N_GRAPHS / 128, 256, 0, stream>>>(yflat, lin1_w, lin1_b, z1, LIN1_IN);
    lin2_lsm<<<2, 256, 0, stream>>>(z1, lin2_w, lin2_b, out);
}